// Correlation_5806795784849
// MI455X (gfx1250) — compile-verified
//
#include <hip/hip_runtime.h>

typedef float v2f __attribute__((ext_vector_type(2)));
typedef float v8f __attribute__((ext_vector_type(8)));

#define CC 256
#define HH 96
#define WW 160
#define BB 8
#define KCH 32                    // channels per K-chunk
#define A_STRIDE 34               // even pad -> 8B-aligned v2f loads, spread banks
#define B_STRIDE 34
#define A_OFF 0
#define B_OFF (16 * A_STRIDE)                    // 544 floats
#define SMEM_FLOATS (B_OFF + 9 * 32 * B_STRIDE)  // 544 + 9792 = 10336 floats (~41 KB)

// out[b, dy*9+dx, y, x] = (1/C) * sum_c in1[b,c,y,x] * in2[b,c,y+dy-4,x+dx-4]
// One block: 16-wide x-tile at (b, y); 9 waves, wave w handles displacement row dy=w.
// Banded-GEMM: Cfull[16 x 32] = A[16 x C] * B[C x 32] via v_wmma_f32_16x16x4_f32,
// then out[m, dx] = Cfull[m][m + dx] (band extraction through LDS).
// Staging is float2-vectorized and register-double-buffered: chunk i+1's global
// loads are in flight while chunk i's WMMAs run.
__global__ __launch_bounds__(288)
void corr_wmma_kernel(const float* __restrict__ in1,
                      const float* __restrict__ in2,
                      float* __restrict__ out)
{
    __shared__ __align__(16) float smem[SMEM_FLOATS];

    const int x0 = blockIdx.x * 16;
    const int y  = blockIdx.y;
    const int b  = blockIdx.z;

    const int tid  = threadIdx.x;
    const int lane = tid & 31;
    const int wid  = tid >> 5;   // dy index 0..8
    const int hi   = lane >> 4;  // K-half select for f32 16x16x4 operand layout
    const int lm   = lane & 15;  // row (A) / column (B) index

    const int HW = HH * WW;
    const size_t in1b = (size_t)b * CC * HW + (size_t)y * WW + x0;
    const size_t in2b = (size_t)b * CC * HW;

    // Per-lane LDS fragment bases (float indices); all 8-byte aligned.
    const int aBase  = A_OFF + lm * A_STRIDE + 2 * hi;
    const int b0Base = B_OFF + (wid * 32 + lm) * B_STRIDE + 2 * hi;
    const int b1Base = b0Base + 16 * B_STRIDE;

    // A staging slot (threads 0..255 only; whole-wave uniform for wave 8)
    const int aK = tid >> 3;         // channel within chunk
    const int aM = (tid & 7) << 1;   // even m pair

    float2 bReg[16];
    float2 aReg = make_float2(0.0f, 0.0f);

    // Issue global loads for one K-chunk into registers (no LDS traffic here).
    auto loadChunk = [&](int c0) {
        if (tid < 256) {
            aReg = *(const float2*)&in1[in1b + (size_t)(c0 + aK) * HW + aM];
        }
#pragma unroll
        for (int j = 0; j < 16; ++j) {
            const int p   = tid + 288 * j;   // pair index, 4608 total = 288*16
            const int dyi = p >> 9;          // 0..8
            const int rem = p & 511;
            const int k   = rem >> 4;        // 0..31
            const int cp  = rem & 15;        // column pair 0..15 (pairs >=12 are pad)
            const int yy  = y + dyi - 4;
            const int xx  = x0 + (cp << 1) - 4;  // even; pairs never straddle bounds
            float2 v = make_float2(0.0f, 0.0f);
            if (cp < 12 && (unsigned)yy < (unsigned)HH && (unsigned)xx < (unsigned)(WW - 1)) {
                v = *(const float2*)&in2[in2b + (size_t)(c0 + k) * HW + (size_t)yy * WW + xx];
            }
            bReg[j] = v;
        }
        // L2 prefetch one further chunk ahead: one (dy,k) row per thread (288 rows).
        {
            const int pdy = tid >> 5;   // 0..8
            const int pk  = tid & 31;
            const int pyy = y + pdy - 4;
            const int pxx = (x0 >= 4) ? (x0 - 4) : 0;
            if (c0 + KCH < CC && (unsigned)pyy < (unsigned)HH) {
                __builtin_prefetch(
                    &in2[in2b + (size_t)(c0 + KCH + pk) * HW + (size_t)pyy * WW + pxx], 0, 1);
            }
        }
    };

    // Drain staged registers into LDS (A[m][k], B[dy][col][k]; cols 24..31 zeroed).
    auto storeChunk = [&]() {
        if (tid < 256) {
            smem[A_OFF + aM * A_STRIDE + aK]       = aReg.x;
            smem[A_OFF + (aM + 1) * A_STRIDE + aK] = aReg.y;
        }
#pragma unroll
        for (int j = 0; j < 16; ++j) {
            const int p    = tid + 288 * j;
            const int dyi  = p >> 9;
            const int rem  = p & 511;
            const int k    = rem >> 4;
            const int col  = (rem & 15) << 1;
            const int base = B_OFF + (dyi * 32 + col) * B_STRIDE + k;
            smem[base]            = bReg[j].x;
            smem[base + B_STRIDE] = bReg[j].y;
        }
    };

    v8f acc0 = {};  // Cfull columns 0..15
    v8f acc1 = {};  // Cfull columns 16..31 (cols 24..31 zero-padded garbage)

    loadChunk(0);

    for (int ci = 0; ci < CC / KCH; ++ci) {
        __syncthreads();                 // all waves done reading LDS (prev chunk)
        storeChunk();                    // implicit wait on this chunk's loads
        if (ci + 1 < CC / KCH) {
            loadChunk((ci + 1) * KCH);   // next chunk's loads fly during WMMAs
        }
        __syncthreads();                 // staged chunk visible to all waves

#pragma unroll
        for (int kk = 0; kk < KCH; kk += 4) {
            v2f a  = *(const v2f*)&smem[aBase  + kk];
            v2f b0 = *(const v2f*)&smem[b0Base + kk];
            v2f b1 = *(const v2f*)&smem[b1Base + kk];
            acc0 = __builtin_amdgcn_wmma_f32_16x16x4_f32(
                false, a, false, b0, (short)0, acc0, false, false);
            acc1 = __builtin_amdgcn_wmma_f32_16x16x4_f32(
                false, a, false, b1, (short)0, acc1, false, false);
        }
    }

    // ---- epilogue: band extraction. Reuse B region as Cbuf[9][16][33]. ----
    __syncthreads();
    const int cBase = B_OFF + (wid * 16) * 33;
    // C/D layout: VGPR r, lanes 0-15 -> M=r, N=lane; lanes 16-31 -> M=r+8, N=lane-16
#pragma unroll
    for (int r = 0; r < 8; ++r) {
        smem[cBase + (r + 8 * hi) * 33 + lm]      = acc0[r];
        smem[cBase + (r + 8 * hi) * 33 + 16 + lm] = acc1[r];
    }
    __syncthreads();
    // out[b, wid*9 + d, y, x0 + m] = Cfull[m][m + d] / 256 ; 16-wide contiguous stores
    for (int i = lane; i < 144; i += 32) {
        int m = i & 15;
        int d = i >> 4;  // 0..8
        float v = smem[cBase + m * 33 + m + d] * (1.0f / 256.0f);
        out[(((size_t)b * 81 + wid * 9 + d) * HH + y) * WW + x0 + m] = v;
    }
}

extern "C" void kernel_launch(void* const* d_in, const int* in_sizes, int n_in,
                              void* d_out, int out_size, void* d_ws, size_t ws_size,
                              hipStream_t stream) {
    const float* in1 = (const float*)d_in[0];
    const float* in2 = (const float*)d_in[1];
    float* out = (float*)d_out;
    dim3 grid(WW / 16, HH, BB);  // (10, 96, 8) blocks, 288 threads = 9 waves
    corr_wmma_kernel<<<grid, dim3(288), 0, stream>>>(in1, in2, out);
}